// Net_5918464934508
// MI455X (gfx1250) — compile-verified
//
#include <hip/hip_runtime.h>
#include <math.h>

typedef float v2f __attribute__((ext_vector_type(2)));
typedef float v8f __attribute__((ext_vector_type(8)));

#define Bb   16
#define Nn   128
#define FIN  8
#define Ss   8
#define FOUT 64
#define KH   32
#define KJ   272   // 256 (k*f) + 8 (kn_b2 path via AX) + 8 (root_w path via X)

// ---------------------------------------------------------------------------
// K0: pack W2' [272 x 64]:  j<256 -> kn_w2[k=j/8][ (j%8)*64+o ]
//                           256..263 -> kn_b2[(j-256)*64+o]
//                           264..271 -> root_w[(j-264)*64+o]
// ---------------------------------------------------------------------------
__global__ void k0_pack_w2(const float* __restrict__ w2,
                           const float* __restrict__ b2,
                           const float* __restrict__ rootw,
                           float* __restrict__ W2p) {
    int idx = blockIdx.x * blockDim.x + threadIdx.x;
    if (idx >= KJ * FOUT) return;
    int j = idx >> 6;
    int o = idx & 63;
    float v;
    if (j < 256)      v = w2[(j >> 3) * (FIN * FOUT) + (j & 7) * FOUT + o];
    else if (j < 264) v = b2[(j - 256) * FOUT + o];
    else              v = rootw[(j - 264) * FOUT + o];
    W2p[idx] = v;
}

// ---------------------------------------------------------------------------
// K1: one wave per (b,n). Computes h = relu(E.w1+b1) per edge, masks by A,
// and accumulates G[b,n,k,f] = sum_i (A*h)[i,k] * X[b,i,f] with
// V_WMMA_F32_16X16X4_F32 (M=32 via two tiles, N=16 with f<8 valid, K=i).
// Also appends AX row (bias path) and X row (root path) into G.
// ---------------------------------------------------------------------------
__global__ __launch_bounds__(32) void k1_edge_gemm(
    const float* __restrict__ A, const float* __restrict__ X,
    const float* __restrict__ E, const float* __restrict__ w1,
    const float* __restrict__ b1, float* __restrict__ G)
{
    const int lane = threadIdx.x;           // 0..31 (wave32)
    const int bn   = blockIdx.x;            // 0..2047
    const int b    = bn >> 7;
    const int n    = bn & 127;

    // per-lane column of layer-1 weights: h_k uses kn_w1[:,k], k = lane
    float w1c[FIN];
#pragma unroll
    for (int f = 0; f < FIN; ++f) w1c[f] = w1[f * KH + lane];
    const float b1c = b1[lane];

    const float* Arow = A + (size_t)bn * Nn;        // A[b,n,:]
    const float* Eb   = E + (size_t)bn * Nn * Ss;   // E[b,n,:,:]
    const float* Xb   = X + (size_t)b  * Nn * FIN;  // X[b,:,:]

    v8f c0 = {0.f,0.f,0.f,0.f,0.f,0.f,0.f,0.f};
    v8f c1 = {0.f,0.f,0.f,0.f,0.f,0.f,0.f,0.f};

    const int  col    = lane & 15;
    const int  khalf  = (lane >> 4) << 1;   // 0 (lanes 0-15) or 2 (lanes 16-31)
    const bool lowhlf = (lane < 16);

    for (int i0 = 0; i0 < Nn; i0 += 4) {
        // h[j][lane] = A[b,n,i0+j] * relu(E[b,n,i0+j,:] . w1[:,lane] + b1[lane])
        float h[4];
#pragma unroll
        for (int j = 0; j < 4; ++j) {
            const float* e = Eb + (size_t)(i0 + j) * Ss;
            float acc = b1c;
#pragma unroll
            for (int f = 0; f < FIN; ++f) acc += e[f] * w1c[f];
            acc = acc > 0.f ? acc : 0.f;
            h[j] = acc * Arow[i0 + j];
        }
        // 16x4 f32 A-fragment layout (ISA 7.12.2):
        //   lanes 0-15 : M=lane,    regs = {K0,K1}
        //   lanes16-31 : M=lane-16, regs = {K2,K3}
        float h0x = __shfl_xor(h[0], 16, 32);
        float h1x = __shfl_xor(h[1], 16, 32);
        float h2x = __shfl_xor(h[2], 16, 32);
        float h3x = __shfl_xor(h[3], 16, 32);
        v2f a0, a1;
        a0.x = lowhlf ? h[0] : h2x;   // M-tile 0: rows k=0..15
        a0.y = lowhlf ? h[1] : h3x;
        a1.x = lowhlf ? h0x  : h[2];  // M-tile 1: rows k=16..31
        a1.y = lowhlf ? h1x  : h[3];

        // 4x16 B-fragment: B[kk][n] = X[b, i0+kk, n] (n<8 valid, else 0)
        v2f bf;
        bf.x = (col < FIN) ? Xb[(size_t)(i0 + khalf    ) * FIN + col] : 0.f;
        bf.y = (col < FIN) ? Xb[(size_t)(i0 + khalf + 1) * FIN + col] : 0.f;

        c0 = __builtin_amdgcn_wmma_f32_16x16x4_f32(false, a0, false, bf,
                                                   (short)0, c0, false, false);
        c1 = __builtin_amdgcn_wmma_f32_16x16x4_f32(false, a1, false, bf,
                                                   (short)0, c1, false, false);
    }

    float* Grow = G + (size_t)bn * KJ;

    // AX[f] = sum_i A[b,n,i] * X[b,i,f]  (bias path), all 32 lanes cooperate
    {
        int f = lane & 7, q = lane >> 3;
        float ax = 0.f;
        for (int i = q; i < Nn; i += 4) ax += Arow[i] * Xb[(size_t)i * FIN + f];
        ax += __shfl_xor(ax, 8, 32);
        ax += __shfl_xor(ax, 16, 32);
        if (lane < 8) {
            Grow[256 + lane] = ax;                              // bias path
            Grow[264 + lane] = Xb[(size_t)n * FIN + lane];      // root path
        }
    }

    // C layout (ISA 7.12.2): VGPR v -> row M=v (lanes 0-15) / M=v+8 (16-31)
    if (col < FIN) {
        const int hi = (lane >> 4) * 8;
#pragma unroll
        for (int v = 0; v < 8; ++v) {
            Grow[(v + hi) * FIN + col]        = c0[v];   // k = 0..15
            Grow[(16 + v + hi) * FIN + col]   = c1[v];   // k = 16..31
        }
    }
}

// ---------------------------------------------------------------------------
// K2: Y = relu(G[2048x272] @ W2'[272x64] + conv_b). 4 waves/block, each wave
// owns a 16x64 output tile (4 accumulators), 68 K-steps of WMMA f32 16x16x4.
// ---------------------------------------------------------------------------
__global__ __launch_bounds__(128) void k2_gemm_relu(
    const float* __restrict__ G, const float* __restrict__ W2p,
    const float* __restrict__ convb, float* __restrict__ Y)
{
    const int lane = threadIdx.x & 31;
    const int wave = threadIdx.x >> 5;
    const int m0   = (blockIdx.x * 4 + wave) * 16;   // row tile base, 0..2032

    v8f c[4];
#pragma unroll
    for (int nt = 0; nt < 4; ++nt) c[nt] = (v8f){0.f,0.f,0.f,0.f,0.f,0.f,0.f,0.f};

    const int col   = lane & 15;
    const int khalf = (lane >> 4) << 1;
    const float* Ar = G + (size_t)(m0 + col) * KJ;

    for (int k0 = 0; k0 < KJ; k0 += 4) {
        v2f a;
        a.x = Ar[k0 + khalf];
        a.y = Ar[k0 + khalf + 1];
#pragma unroll
        for (int nt = 0; nt < 4; ++nt) {
            const float* Bp = W2p + (size_t)(k0 + khalf) * FOUT + nt * 16 + col;
            v2f bf;
            bf.x = Bp[0];
            bf.y = Bp[FOUT];
            c[nt] = __builtin_amdgcn_wmma_f32_16x16x4_f32(false, a, false, bf,
                                                          (short)0, c[nt], false, false);
        }
    }

    const int rowoff = (lane >> 4) * 8;
#pragma unroll
    for (int nt = 0; nt < 4; ++nt) {
        const int o    = nt * 16 + col;
        const float bv = convb[o];
#pragma unroll
        for (int v = 0; v < 8; ++v) {
            float val = c[nt][v] + bv;
            val = val > 0.f ? val : 0.f;
            Y[(size_t)(m0 + rowoff + v) * FOUT + o] = val;
        }
    }
}

// ---------------------------------------------------------------------------
// K3: per-batch mean-pool over nodes + Dense(32,relu) + Dense(1,sigmoid)
// ---------------------------------------------------------------------------
__global__ __launch_bounds__(64) void k3_head(
    const float* __restrict__ Y, const float* __restrict__ fcw,
    const float* __restrict__ fcb, const float* __restrict__ ow,
    const float* __restrict__ ob, float* __restrict__ out)
{
    __shared__ float y[FOUT];
    __shared__ float t[32];
    const int b   = blockIdx.x;
    const int tid = threadIdx.x;

    float s = 0.f;
    for (int n = 0; n < Nn; ++n) s += Y[((size_t)b * Nn + n) * FOUT + tid];
    y[tid] = s * (1.0f / (float)Nn);
    __syncthreads();

    if (tid < 32) {
        float acc = fcb[tid];
        for (int o = 0; o < FOUT; ++o) acc += y[o] * fcw[o * 32 + tid];
        t[tid] = acc > 0.f ? acc : 0.f;
    }
    __syncthreads();

    if (tid == 0) {
        float z = ob[0];
        for (int j = 0; j < 32; ++j) z += t[j] * ow[j];
        out[b] = 1.0f / (1.0f + expf(-z));
    }
}

// ---------------------------------------------------------------------------
extern "C" void kernel_launch(void* const* d_in, const int* in_sizes, int n_in,
                              void* d_out, int out_size, void* d_ws, size_t ws_size,
                              hipStream_t stream) {
    const float* A     = (const float*)d_in[0];
    const float* X     = (const float*)d_in[1];
    const float* E     = (const float*)d_in[2];
    const float* kn_w1 = (const float*)d_in[3];
    const float* kn_b1 = (const float*)d_in[4];
    const float* kn_w2 = (const float*)d_in[5];
    const float* kn_b2 = (const float*)d_in[6];
    const float* rootw = (const float*)d_in[7];
    const float* convb = (const float*)d_in[8];
    const float* fcw   = (const float*)d_in[9];
    const float* fcb   = (const float*)d_in[10];
    const float* ow    = (const float*)d_in[11];
    const float* ob    = (const float*)d_in[12];
    float* out = (float*)d_out;

    char* ws = (char*)d_ws;
    float* W2p = (float*)(ws);                               // 272*64*4   = 69,632 B
    float* G   = (float*)(ws + 69632);                       // 2048*272*4 = 2,228,224 B
    float* Y   = (float*)(ws + 69632 + 2228224);             // 2048*64*4  = 524,288 B

    k0_pack_w2<<<(KJ * FOUT + 255) / 256, 256, 0, stream>>>(kn_w2, kn_b2, rootw, W2p);
    k1_edge_gemm<<<Bb * Nn, 32, 0, stream>>>(A, X, E, kn_w1, kn_b1, G);
    k2_gemm_relu<<<32, 128, 0, stream>>>(G, W2p, convb, Y);
    k3_head<<<Bb, 64, 0, stream>>>(Y, fcw, fcb, ow, ob, out);
}